// MyGraphConvolution_87832081203638
// MI455X (gfx1250) — compile-verified
//
#include <hip/hip_runtime.h>
#include <hip/hip_bf16.h>
#include <math.h>

// Shapes: C=64, K=5, B=64, T=2000
#define GC_C 64
#define GC_K 5
#define GC_B 64
#define GC_T 2000
#define JACOBI_SWEEPS 16

typedef __attribute__((ext_vector_type(2))) float v2f;
typedef __attribute__((ext_vector_type(8))) float v8f;

// ---------------------------------------------------------------------------
// Kernel 1: build L = diag(colsum(A)) - A, Jacobi eigensolver (parallel cyclic
// tournament schedule), rank eigenvalues ascending, emit
// kern[k][i][j] = sum_u V[i][u] * g[k][rank[u]] * V[j][u].
// Single workgroup, 256 threads. All in LDS (padded stride 65).
// ---------------------------------------------------------------------------
__device__ __forceinline__ int jac_player(int pos, int round) {
    // circle-method round robin: player 0 pinned at position 0,
    // players 1..63 rotate through positions 1..63.
    return (pos == 0) ? 0 : (((pos - 1 + round) % 63) + 1);
}

__global__ __launch_bounds__(256)
void gc_eig_kern(const float* __restrict__ adj,     // 64x64
                 const float* __restrict__ gker,    // 5x1x64
                 float* __restrict__ kern_out) {    // 5x64x64
    __shared__ float Am[64][65];
    __shared__ float Vm[64][65];
    __shared__ float cs_c[32], cs_s[32];
    __shared__ int   pp[32], qq[32];
    __shared__ float wv[64];
    __shared__ int   rnk[64];
    __shared__ float geff[GC_K * 64];

    const int tid = threadIdx.x;

    // ---- L = diag(colsum(A)) - A ; V = I ----
    for (int idx = tid; idx < 64 * 64; idx += 256) {
        int i = idx >> 6, j = idx & 63;
        Am[i][j] = -adj[idx];
        Vm[i][j] = (i == j) ? 1.0f : 0.0f;
    }
    __syncthreads();
    if (tid < 64) {
        float s = 0.0f;
        for (int j = 0; j < 64; ++j) s += adj[j * 64 + tid];  // colsum
        Am[tid][tid] += s;
    }
    __syncthreads();

    // ---- parallel cyclic Jacobi ----
    for (int sweep = 0; sweep < JACOBI_SWEEPS; ++sweep) {
        for (int round = 0; round < 63; ++round) {
            if (tid < 32) {
                int ia = jac_player(tid, round);
                int ib = jac_player(63 - tid, round);
                int p = min(ia, ib), q = max(ia, ib);
                pp[tid] = p; qq[tid] = q;
                float app = Am[p][p], aqq = Am[q][q], apq = Am[p][q];
                float cc = 1.0f, ss = 0.0f;
                if (fabsf(apq) > 1e-30f) {
                    float tau = (aqq - app) / (2.0f * apq);
                    float t = copysignf(1.0f, tau) /
                              (fabsf(tau) + sqrtf(1.0f + tau * tau));
                    cc = 1.0f / sqrtf(1.0f + t * t);
                    ss = t * cc;
                }
                cs_c[tid] = cc; cs_s[tid] = ss;
            }
            __syncthreads();
            // A <- A*J (columns), V <- V*J
            for (int idx = tid; idx < 2048; idx += 256) {
                int r = idx >> 5, j = idx & 31;
                int p = pp[j], q = qq[j];
                float cc = cs_c[j], ss = cs_s[j];
                float ap = Am[r][p], aq = Am[r][q];
                Am[r][p] = cc * ap - ss * aq;
                Am[r][q] = ss * ap + cc * aq;
                float vp = Vm[r][p], vq = Vm[r][q];
                Vm[r][p] = cc * vp - ss * vq;
                Vm[r][q] = ss * vp + cc * vq;
            }
            __syncthreads();
            // A <- J^T * A (rows)
            for (int idx = tid; idx < 2048; idx += 256) {
                int c = idx >> 5, j = idx & 31;
                int p = pp[j], q = qq[j];
                float cc = cs_c[j], ss = cs_s[j];
                float ap = Am[p][c], aq = Am[q][c];
                Am[p][c] = cc * ap - ss * aq;
                Am[q][c] = ss * ap + cc * aq;
            }
            __syncthreads();
        }
    }

    // ---- ascending rank of eigenvalues ----
    if (tid < 64) wv[tid] = Am[tid][tid];
    __syncthreads();
    if (tid < 64) {
        float wu = wv[tid];
        int r = 0;
        for (int j = 0; j < 64; ++j)
            r += (wv[j] < wu) || (wv[j] == wu && j < tid);
        rnk[tid] = r;
    }
    __syncthreads();
    // geff[k][u] = g[k][rank[u]]
    for (int idx = tid; idx < GC_K * 64; idx += 256) {
        int k = idx >> 6, u = idx & 63;
        geff[idx] = gker[k * 64 + rnk[u]];
    }
    __syncthreads();

    // ---- kern[k][i][j] = sum_u V[i][u] * geff[k][u] * V[j][u] ----
    for (int idx = tid; idx < GC_K * 64 * 64; idx += 256) {
        int k = idx >> 12;
        int i = (idx >> 6) & 63;
        int j = idx & 63;
        const float* ge = &geff[k * 64];
        float sum = 0.0f;
        #pragma unroll 8
        for (int u = 0; u < 64; ++u)
            sum += Vm[i][u] * Vm[j][u] * ge[u];
        kern_out[idx] = sum;
    }
}

// ---------------------------------------------------------------------------
// Kernel 2: out[b,k] = relu(kern[k] @ x[b,k]) via V_WMMA_F32_16X16X4_F32.
// Wave owns full M=64 strip of one 16-wide T chunk; kern[k] cached as 64
// A-fragments in registers, reused across chunks. EXEC all-1s at every WMMA.
// Grid: x = b*K+k (320), y = T segment (4). Block: 256 threads = 8 waves.
// ---------------------------------------------------------------------------
#define T_CHUNKS (GC_T / 16)   // 125

__global__ __launch_bounds__(256)
void gc_matmul(const float* __restrict__ x,      // B,K,C,T
               const float* __restrict__ kern,   // K,C,C
               float* __restrict__ out) {        // B,K,C,T
    const int bk   = blockIdx.x;          // 0..319
    const int k    = bk % GC_K;
    const int wave = threadIdx.x >> 5;
    const int lane = threadIdx.x & 31;
    const int lo   = lane & 15;           // N column within tile / M row within A frag
    const int hi   = lane >> 4;           // half-wave select

    const float* __restrict__ kb = kern + (size_t)k * 64 * 64;
    const float* __restrict__ xb = x    + (size_t)bk * 64 * GC_T;
    float*       __restrict__ ob = out  + (size_t)bk * 64 * GC_T;

    // Preload kern[k] as WMMA A fragments (16x4 f32 layout):
    // lanes 0-15: M=lo, K={0,1}; lanes 16-31: M=lo, K={2,3}.
    v2f af[4][16];
    #pragma unroll
    for (int m = 0; m < 4; ++m) {
        #pragma unroll
        for (int s = 0; s < 16; ++s) {
            const float* p = kb + (m * 16 + lo) * 64 + 4 * s + 2 * hi;
            af[m][s].x = p[0];
            af[m][s].y = p[1];
        }
    }

    const int cstride = 8 * gridDim.y;    // waves per bk across all y-blocks
    for (int chunk = wave + 8 * blockIdx.y; chunk < T_CHUNKS; chunk += cstride) {
        const int t0 = chunk * 16;

        // prefetch next chunk's x tile (one lane pair per K-row)
        const int tn = t0 + cstride * 16;
        if (tn < GC_T) {
            #pragma unroll 4
            for (int r = 0; r < 4; ++r)
                __builtin_prefetch(&xb[(r * 16 + lane / 2) * GC_T + tn + (lane & 1) * 8], 0, 3);
        }

        v8f acc[4] = {};
        #pragma unroll
        for (int s = 0; s < 16; ++s) {
            const int kk = 4 * s;
            // B fragment (4x16 f32): VGPR v, half h -> row K = v + 2h, col = lo
            v2f bf;
            bf.x = xb[(kk + 2 * hi    ) * GC_T + t0 + lo];
            bf.y = xb[(kk + 2 * hi + 1) * GC_T + t0 + lo];
            #pragma unroll
            for (int m = 0; m < 4; ++m) {
                acc[m] = __builtin_amdgcn_wmma_f32_16x16x4_f32(
                    false, af[m][s], false, bf, (short)0, acc[m], false, false);
            }
        }

        // D tile (16x16 f32): VGPR v, half h -> row M = v + 8h, col = lo.  Fused ReLU.
        #pragma unroll
        for (int m = 0; m < 4; ++m) {
            #pragma unroll
            for (int v = 0; v < 8; ++v) {
                const int row = m * 16 + v + 8 * hi;
                ob[row * GC_T + t0 + lo] = fmaxf(acc[m][v], 0.0f);
            }
        }
    }
}

// ---------------------------------------------------------------------------
extern "C" void kernel_launch(void* const* d_in, const int* in_sizes, int n_in,
                              void* d_out, int out_size, void* d_ws, size_t ws_size,
                              hipStream_t stream) {
    const float* x    = (const float*)d_in[0];   // (B,K,C,T)
    const float* adj  = (const float*)d_in[1];   // (C,C)
    const float* gker = (const float*)d_in[2];   // (K,1,C)
    float* out  = (float*)d_out;
    float* kern = (float*)d_ws;                  // K*C*C floats = 80 KB scratch

    gc_eig_kern<<<1, 256, 0, stream>>>(adj, gker, kern);
    gc_matmul<<<dim3(GC_B * GC_K, 4), 256, 0, stream>>>(x, kern, out);
}